// CRF_27925877358592
// MI455X (gfx1250) — compile-verified
//
#include <hip/hip_runtime.h>
#include <stdint.h>

// ---------------------------------------------------------------------------
// CRF forward (score - logZ), B=8192, S=1024, NT=4, wave32 / gfx1250.
// One lane per batch element. Emissions/tags/mask staged through LDS with
// double-buffered global_load_async_to_lds_* (ASYNCcnt) per CDNA5 ISA ch.10.
// ---------------------------------------------------------------------------

#define NEGV (-10000.0f)
#define L2E  1.4426950408889634f
#define LN2  0.6931471805599453f

constexpr int B_  = 8192;
constexpr int S_  = 1024;
constexpr int NT_ = 4;

constexpr int TB = 128;            // threads per block == batch rows per block
constexpr int TS = 8;              // timesteps per tile
constexpr int NTILES = S_ / TS;    // 128

constexpr int EROW = TS * 16 + 16; // 144 B: 8 x float4 + pad (bank stagger, 16B aligned)
constexpr int TROW = TS * 4 + 16;  // 48 B : 8 x int32  + pad (16B aligned for b128)
constexpr int MROW = 16;           // 8 mask bytes + pad (8B aligned for b64)
constexpr int E_BYTES = TB * EROW; // 18432
constexpr int T_BYTES = TB * TROW; // 6144
constexpr int M_BYTES = TB * MROW; // 2048
constexpr int SMEM_BYTES = 2 * E_BYTES + 2 * T_BYTES + 2 * M_BYTES; // 53248 < 64KB

__device__ __forceinline__ float fast_exp2(float x) {
#if defined(__HIP_DEVICE_COMPILE__) && __has_builtin(__builtin_amdgcn_exp2f)
  return __builtin_amdgcn_exp2f(x);     // v_exp_f32
#else
  return exp2f(x);
#endif
}
__device__ __forceinline__ float fast_log2(float x) {
#if defined(__HIP_DEVICE_COMPILE__) && __has_builtin(__builtin_amdgcn_logf)
  return __builtin_amdgcn_logf(x);      // v_log_f32
#else
  return log2f(x);
#endif
}

// log(exp(x)+exp(y)) with 1 exp + 1 log (hardware base-2 trans ops).
__device__ __forceinline__ float lse2(float x, float y) {
  float mx = fmaxf(x, y);
  float d  = fminf(x, y) - mx;                       // <= 0; underflows to 0 for -1e4 paths
  return mx + LN2 * fast_log2(1.0f + fast_exp2(d * L2E));
}

__device__ __forceinline__ float sel4(float4 e, int c) {
  float r = (c == 0) ? e.x : (c == 1) ? e.y : (c == 2) ? e.z : e.w;
  return r;
}

// Issue one tile's async global->LDS copies (11 wave-level async ops).
__device__ __forceinline__ void issue_async_tile(
    const float* __restrict__ emis, const int* __restrict__ tags,
    const unsigned char* __restrict__ mask, int b, int t0,
    unsigned ldsE, unsigned ldsT, unsigned ldsM) {
  unsigned long long gE =
      (unsigned long long)(emis + ((size_t)b * S_ + (size_t)t0) * NT_);
  unsigned long long gT =
      (unsigned long long)(tags + ((size_t)b * S_ + (size_t)t0));
  unsigned long long gM =
      (unsigned long long)(mask + ((size_t)b * S_ + (size_t)t0));
#pragma unroll
  for (int k = 0; k < TS; ++k) {
    asm volatile("global_load_async_to_lds_b128 %0, %1, off"
                 :: "v"(ldsE + (unsigned)(k * 16)),
                    "v"(gE + (unsigned long long)(k * 16))
                 : "memory");
  }
  asm volatile("global_load_async_to_lds_b128 %0, %1, off"
               :: "v"(ldsT), "v"(gT) : "memory");
  asm volatile("global_load_async_to_lds_b128 %0, %1, off"
               :: "v"(ldsT + 16u), "v"(gT + 16ull) : "memory");
  asm volatile("global_load_async_to_lds_b64 %0, %1, off"
               :: "v"(ldsM), "v"(gM) : "memory");
}

__global__ __launch_bounds__(TB) void crf_fwd_kernel(
    const float* __restrict__ emissions,
    const float* __restrict__ transitions,
    const float* __restrict__ start_tr,
    const float* __restrict__ end_tr,
    const int* __restrict__ tags,
    const unsigned char* __restrict__ mask,
    float* __restrict__ out) {
  __shared__ alignas(16) unsigned char smem[SMEM_BYTES];

  const int tid = threadIdx.x;
  const int b   = blockIdx.x * TB + tid;

  // Masked parameters (uniform -> scalar loads). Mask structure is constant:
  // rows {0,1} -> cols {1,2}; rows {2,3} -> cols {0,3}; start {0,3}; end {2,3}.
  const float t01 = transitions[0 * 4 + 1], t02 = transitions[0 * 4 + 2];
  const float t11 = transitions[1 * 4 + 1], t12 = transitions[1 * 4 + 2];
  const float t20 = transitions[2 * 4 + 0], t23 = transitions[2 * 4 + 3];
  const float t30 = transitions[3 * 4 + 0], t33 = transitions[3 * 4 + 3];
  const float st0 = start_tr[0], st3 = start_tr[3];
  const float et2 = end_tr[2],   et3 = end_tr[3];

  const unsigned ldsBase = (unsigned)(unsigned long long)(void*)smem;
  const int eoff[2] = {0, E_BYTES};
  const int toff[2] = {2 * E_BYTES, 2 * E_BYTES + T_BYTES};
  const int moff[2] = {2 * E_BYTES + 2 * T_BYTES,
                       2 * E_BYTES + 2 * T_BYTES + M_BYTES};

  float a0 = 0.f, a1 = 0.f, a2 = 0.f, a3 = 0.f, score = 0.f;
  int prev = 0, last = 0;

  // Prime the pipeline: tile 0 -> buffer 0.
  issue_async_tile(emissions, tags, mask, b, 0,
                   ldsBase + (unsigned)(eoff[0] + tid * EROW),
                   ldsBase + (unsigned)(toff[0] + tid * TROW),
                   ldsBase + (unsigned)(moff[0] + tid * MROW));

  for (int tile = 0; tile < NTILES; ++tile) {
    const int buf = tile & 1;
    if (tile + 1 < NTILES) {
      const int nb = buf ^ 1;  // safe: prior compute on nb finished + barrier'd
      issue_async_tile(emissions, tags, mask, b, (tile + 1) * TS,
                       ldsBase + (unsigned)(eoff[nb] + tid * EROW),
                       ldsBase + (unsigned)(toff[nb] + tid * TROW),
                       ldsBase + (unsigned)(moff[nb] + tid * MROW));
      // 11 newer async ops in flight; in-order completion => <=11 drains tile.
      asm volatile("s_wait_asynccnt 0xb" ::: "memory");
    } else {
      asm volatile("s_wait_asynccnt 0x0" ::: "memory");
    }
    __syncthreads();  // all waves' tile data resident in LDS

    const unsigned char* eRow = smem + eoff[buf] + tid * EROW;
    const int* tRow = (const int*)(smem + toff[buf] + tid * TROW);
    const unsigned char* mRow = smem + moff[buf] + tid * MROW;

#pragma unroll
    for (int tl = 0; tl < TS; ++tl) {
      float4 e = *(const float4*)(eRow + tl * 16);  // ds_load_b128
      int c = tRow[tl];
      unsigned char mv = mRow[tl];

      if (tile == 0 && tl == 0) {
        float stc = (c == 0) ? st0 : (c == 3) ? st3 : NEGV;
        score = stc + sel4(e, c);
        a0 = st0 + e.x;  a1 = NEGV + e.y;
        a2 = NEGV + e.z; a3 = st3 + e.w;
        prev = c; last = c;
      } else {
        const bool m = (mv != 0);
        const float fm = m ? 1.0f : 0.0f;
        // structured transition lookup (7 selects instead of 16-way table)
        float tsel;
        if (prev < 2) {
          float r1 = (prev == 0) ? t01 : t11;
          float r2 = (prev == 0) ? t02 : t12;
          tsel = (c == 1) ? r1 : (c == 2) ? r2 : NEGV;
        } else {
          float r0 = (prev == 2) ? t20 : t30;
          float r3 = (prev == 2) ? t23 : t33;
          tsel = (c == 0) ? r0 : (c == 3) ? r3 : NEGV;
        }
        score = fmaf(tsel + sel4(e, c), fm, score);
        // forward recursion: only 2 legal predecessors per state
        float n0 = lse2(a2 + t20, a3 + t30) + e.x;
        float n1 = lse2(a0 + t01, a1 + t11) + e.y;
        float n2 = lse2(a0 + t02, a1 + t12) + e.z;
        float n3 = lse2(a2 + t23, a3 + t33) + e.w;
        a0 = m ? n0 : a0; a1 = m ? n1 : a1;
        a2 = m ? n2 : a2; a3 = m ? n3 : a3;
        last = m ? c : last;
        prev = c;
      }
    }
    __syncthreads();  // all lanes done reading buf before it is refilled
  }

  // end transitions + logZ
  score += (last == 2) ? et2 : (last == 3) ? et3 : NEGV;
  float z0 = a0 + NEGV, z1 = a1 + NEGV, z2 = a2 + et2, z3 = a3 + et3;
  float mz = fmaxf(fmaxf(z0, z1), fmaxf(z2, z3));
  float s  = fast_exp2((z0 - mz) * L2E) + fast_exp2((z1 - mz) * L2E) +
             fast_exp2((z2 - mz) * L2E) + fast_exp2((z3 - mz) * L2E);
  float logz = mz + LN2 * fast_log2(s);

  out[b] = score - logz;
}

extern "C" void kernel_launch(void* const* d_in, const int* in_sizes, int n_in,
                              void* d_out, int out_size, void* d_ws,
                              size_t ws_size, hipStream_t stream) {
  (void)in_sizes; (void)n_in; (void)out_size; (void)d_ws; (void)ws_size;
  const float* emissions          = (const float*)d_in[0];
  const float* transitions        = (const float*)d_in[1];
  const float* start_transitions  = (const float*)d_in[2];
  const float* end_transitions    = (const float*)d_in[3];
  const int*   tags               = (const int*)d_in[4];
  const unsigned char* mask       = (const unsigned char*)d_in[5];
  float* out = (float*)d_out;

  dim3 grid(B_ / TB), block(TB);
  crf_fwd_kernel<<<grid, block, 0, stream>>>(
      emissions, transitions, start_transitions, end_transitions, tags, mask,
      out);
}